// CharLSTM_55190329753725
// MI455X (gfx1250) — compile-verified
//
#include <hip/hip_runtime.h>
#include <math.h>

#define T_SEQ 4096
#define EMB   512
#define HID   2048
#define HIN   2560
#define NCH   256

#define NBLK  64      // persistent scan blocks (device-wide barrier participants)
#define BTHR  256     // threads per scan block (8 wave32)
#define JPB   (HID / NBLK)   // 32 h-outputs owned per block

typedef float v2f __attribute__((ext_vector_type(2)));
typedef float v8f __attribute__((ext_vector_type(8)));
typedef float f4  __attribute__((ext_vector_type(4)));
typedef int   v4i __attribute__((ext_vector_type(4)));

typedef __attribute__((address_space(1))) v4i* gptr_v4i;
typedef __attribute__((address_space(3))) v4i* lptr_v4i;

#if __has_builtin(__builtin_amdgcn_global_load_async_to_lds_b128) && \
    __has_builtin(__builtin_amdgcn_s_wait_asynccnt)
#define USE_ASYNC_LDS 1
#else
#define USE_ASYNC_LDS 0
#endif

// ---------------------------------------------------------------------------
// Kernel 1: Gtab[g][c][j] = b_g[j] + sum_{k<512} W_g[j][k] * emb[c][k]
// One wave per 16x16 output tile, fp32 WMMA 16x16x4, K-loop of 128.
// ---------------------------------------------------------------------------
__global__ __launch_bounds__(32) void gtab_kernel(
    const float* __restrict__ Wf, const float* __restrict__ bf,
    const float* __restrict__ Wi, const float* __restrict__ bi,
    const float* __restrict__ Wo, const float* __restrict__ bo,
    const float* __restrict__ Wc, const float* __restrict__ bc,
    const float* __restrict__ emb, float* __restrict__ Gtab)
{
    const int m0 = blockIdx.x * 16;         // row (j) tile
    const int n0 = blockIdx.y * 16;         // char tile
    const int g  = blockIdx.z;              // gate
    const float* W = (g == 0) ? Wf : (g == 1) ? Wi : (g == 2) ? Wo : Wc;
    const float* B = (g == 0) ? bf : (g == 1) ? bi : (g == 2) ? bo : bc;

    const int lane = threadIdx.x;
    const int l15  = lane & 15;
    const int koff = (lane >> 4) * 2;

    const float* wrow = W   + (size_t)(m0 + l15) * HIN;   // A: W row-major
    const float* erow = emb + (size_t)(n0 + l15) * EMB;   // B[k][n] = emb[n][k]

    v8f acc = {};
    for (int k = 0; k < EMB; k += 4) {
        v2f a, b;
        a.x = wrow[k + koff]; a.y = wrow[k + koff + 1];
        b.x = erow[k + koff]; b.y = erow[k + koff + 1];
        acc = __builtin_amdgcn_wmma_f32_16x16x4_f32(false, a, false, b,
                                                    (short)0, acc, false, false);
    }

    const int col   = n0 + l15;
    const int mbase = m0 + ((lane >> 4) * 8);
    float* out = Gtab + ((size_t)g * NCH + col) * HID + mbase;   // [g][c][j], j contiguous
#pragma unroll
    for (int r = 0; r < 8; ++r)
        out[r] = acc[r] + B[mbase + r];
}

// ---------------------------------------------------------------------------
// Kernel 2: persistent sequential LSTM scan with device-wide spin barrier.
// Each block owns 32 h-indices; cell state c lives in registers of tid<32.
// Per step: barrier -> h into LDS (async-to-LDS path) -> each wave does 16
// coalesced row-dots of W[:,512:] @ h -> gate math (+Gtab gather) -> write h.
// ---------------------------------------------------------------------------
__global__ __launch_bounds__(BTHR) void scan_kernel(
    const int*   __restrict__ seq,
    const float* __restrict__ Wf, const float* __restrict__ Wi,
    const float* __restrict__ Wo, const float* __restrict__ Wc,
    const float* __restrict__ Gtab,
    float* __restrict__ h_buf, float* __restrict__ h_all,
    unsigned int* __restrict__ counter,
    float* __restrict__ h_last_out, float* __restrict__ c_last_out)
{
    __shared__ float h_lds[HID];
    __shared__ float z_lds[4 * JPB];

    const int tid  = threadIdx.x;
    const int blk  = blockIdx.x;
    const int wave = tid >> 5;
    const int lane = tid & 31;
    const float* Wm[4] = {Wf, Wi, Wo, Wc};

    // initialize h_buf slice to zero (h_0 = 0); c_0 = 0 in registers
    if (tid < JPB) h_buf[blk * JPB + tid] = 0.0f;
    float cstate = 0.0f;

    unsigned int phase = 0;
    for (int t = 0; t < T_SEQ; ++t) {
        // --- device-wide barrier: all h_buf writes of the previous phase visible ---
        ++phase;
        __threadfence();
        __syncthreads();
        if (tid == 0) {
            __hip_atomic_fetch_add(counter, 1u, __ATOMIC_RELEASE, __HIP_MEMORY_SCOPE_AGENT);
            while (__hip_atomic_load(counter, __ATOMIC_ACQUIRE, __HIP_MEMORY_SCOPE_AGENT)
                   < phase * NBLK) {
                __builtin_amdgcn_s_sleep(1);
            }
        }
        __syncthreads();

        // --- stage h into LDS (2048 floats) ---
#if USE_ASYNC_LDS
        // 512 x 16B async copies, no VGPR round-trip; tracked by ASYNCcnt
        for (int i = tid; i < HID / 4; i += BTHR) {
            __builtin_amdgcn_global_load_async_to_lds_b128(
                (gptr_v4i)(h_buf + i * 4),
                (lptr_v4i)(&h_lds[i * 4]),
                0, 0);
        }
        __builtin_amdgcn_s_wait_asynccnt(0);
        __syncthreads();
#else
        for (int i = tid; i < HID; i += BTHR) h_lds[i] = h_buf[i];
        __syncthreads();
#endif

        const int ct = seq[t];

        // --- 16 row-dot-products per wave: z = W_g[j][512:] . h ---
        for (int i = 0; i < 16; ++i) {
            const int rr = wave * 16 + i;          // 0..127 within block
            const int g  = rr >> 5;
            const int jl = rr & 31;
            const int j  = blk * JPB + jl;
            const float* wrow = Wm[g] + (size_t)j * HIN + EMB;
            if (i + 1 < 16) {                      // prefetch next row's slice
                const int r2 = rr + 1;
                const float* nrow = Wm[r2 >> 5] + (size_t)(blk * JPB + (r2 & 31)) * HIN + EMB;
                __builtin_prefetch(nrow + lane * 64, 0, 1);
            }
            float sum = 0.0f;
#pragma unroll 4
            for (int kk = 0; kk < 16; ++kk) {
                const f4 wv = *(const f4*)(wrow + kk * 128 + lane * 4);
                const f4 hv = *(const f4*)(&h_lds[kk * 128 + lane * 4]);
                sum += wv.x * hv.x + wv.y * hv.y + wv.z * hv.z + wv.w * hv.w;
            }
            sum += __shfl_xor(sum, 16);
            sum += __shfl_xor(sum, 8);
            sum += __shfl_xor(sum, 4);
            sum += __shfl_xor(sum, 2);
            sum += __shfl_xor(sum, 1);
            if (lane == 0) z_lds[rr] = sum;
        }
        __syncthreads();

        // --- gate math + state update by the 32 owner threads ---
        if (tid < JPB) {
            const int j = blk * JPB + tid;
            const size_t gbase = (size_t)ct * HID + j;
            const float zf = z_lds[0 * JPB + tid] + Gtab[0 * (size_t)NCH * HID + gbase];
            const float zi = z_lds[1 * JPB + tid] + Gtab[1 * (size_t)NCH * HID + gbase];
            const float zo = z_lds[2 * JPB + tid] + Gtab[2 * (size_t)NCH * HID + gbase];
            const float zc = z_lds[3 * JPB + tid] + Gtab[3 * (size_t)NCH * HID + gbase];
            const float fg = 1.0f / (1.0f + __expf(-zf));
            const float ig = 1.0f / (1.0f + __expf(-zi));
            const float og = 1.0f / (1.0f + __expf(-zo));
            const float gg = tanhf(zc);
            cstate = fg * cstate + ig * gg;
            const float hn = og * tanhf(cstate);
            h_buf[j] = hn;
            h_all[(size_t)t * HID + j] = hn;
            if (t == T_SEQ - 1) { h_last_out[j] = hn; c_last_out[j] = cstate; }
        }
        // next iteration's threadfence+release-add publishes h_buf
    }
}

// ---------------------------------------------------------------------------
// Kernel 3: out_seq[t][n] = bout[n] + sum_k h_all[t][k] * Wout[n][k]
// fp32 WMMA, M = t tiles (256), N = char tiles (16), K = 2048 (512 iters).
// ---------------------------------------------------------------------------
__global__ __launch_bounds__(32) void outgemm_kernel(
    const float* __restrict__ h_all, const float* __restrict__ Wout,
    const float* __restrict__ bout, float* __restrict__ out)
{
    const int m0 = blockIdx.x * 16;      // timestep tile
    const int n0 = blockIdx.y * 16;      // char tile
    const int lane = threadIdx.x;
    const int l15  = lane & 15;
    const int koff = (lane >> 4) * 2;

    const float* arow = h_all + (size_t)(m0 + l15) * HID;   // A = h_all row-major
    const float* brow = Wout  + (size_t)(n0 + l15) * HID;   // B[k][n] = Wout[n][k]

    v8f acc = {};
    for (int k = 0; k < HID; k += 4) {
        v2f a, b;
        a.x = arow[k + koff]; a.y = arow[k + koff + 1];
        b.x = brow[k + koff]; b.y = brow[k + koff + 1];
        acc = __builtin_amdgcn_wmma_f32_16x16x4_f32(false, a, false, b,
                                                    (short)0, acc, false, false);
    }

    const int col   = n0 + l15;
    const int mbase = m0 + ((lane >> 4) * 8);
    const float bb  = bout[col];
#pragma unroll
    for (int r = 0; r < 8; ++r)
        out[(size_t)(mbase + r) * NCH + col] = acc[r] + bb;
}

// ---------------------------------------------------------------------------
extern "C" void kernel_launch(void* const* d_in, const int* in_sizes, int n_in,
                              void* d_out, int out_size, void* d_ws, size_t ws_size,
                              hipStream_t stream)
{
    const int*   seq  = (const int*)  d_in[0];
    const float* emb  = (const float*)d_in[1];
    const float* Wf   = (const float*)d_in[2];  const float* bf = (const float*)d_in[3];
    const float* Wi   = (const float*)d_in[4];  const float* bi = (const float*)d_in[5];
    const float* Wo   = (const float*)d_in[6];  const float* bo = (const float*)d_in[7];
    const float* Wc   = (const float*)d_in[8];  const float* bc = (const float*)d_in[9];
    const float* Wout = (const float*)d_in[10]; const float* bout = (const float*)d_in[11];
    float* out = (float*)d_out;

    char* ws = (char*)d_ws;
    unsigned int* counter = (unsigned int*)ws;                       // 256 B
    float* h_buf = (float*)(ws + 1024);                              // 8 KB
    float* Gtab  = (float*)(ws + (1 << 16));                         // 8 MB: [4][256][2048]
    float* h_all = (float*)(ws + (1 << 16) +
                            (size_t)4 * NCH * HID * sizeof(float));  // 32 MB: [4096][2048]

    (void)hipMemsetAsync(counter, 0, 256, stream);

    gtab_kernel<<<dim3(HID / 16, NCH / 16, 4), 32, 0, stream>>>(
        Wf, bf, Wi, bi, Wo, bo, Wc, bc, emb, Gtab);

    scan_kernel<<<NBLK, BTHR, 0, stream>>>(
        seq, Wf, Wi, Wo, Wc, Gtab, h_buf, h_all, counter,
        out + (size_t)T_SEQ * NCH,              // h_last
        out + (size_t)T_SEQ * NCH + HID);       // c_last

    outgemm_kernel<<<dim3(T_SEQ / 16, NCH / 16), 32, 0, stream>>>(
        h_all, Wout, bout, out);
}